// PNet_Detect_48275432407613
// MI455X (gfx1250) — compile-verified
//
#include <hip/hip_runtime.h>
#include <hip/hip_bf16.h>
#include <stdint.h>

// ---------------------------------------------------------------------------
// PNet detect post-processing for MI455X (gfx1250).
// Single workgroup, everything LDS-resident (CDNA5: 320KB LDS per WGP).
//   Phase 0: TDM tensor_load_to_lds DMA of cls map (96x96 f32) into LDS.
//   Phase 1: build u64 sort keys (score bits | complemented index).
//   Phase 2: bitonic descending sort of 16384 keys in LDS (128 KB).
//   Phase 3: per-thread register cache of boxes + validity (wave32: 9 ranks/thread).
//   Phase 4: greedy NMS, 32-rank tiles resolved inside wave 0 (bitmask greedy),
//            then parallel suppression of later ranks by kept tile members.
//            (Second reference NMS pass @0.7 is provably identity; skipped.)
//   Phase 5: calibrated box output, gated by keep mask.
// ---------------------------------------------------------------------------

#define H_      96
#define W_      96
#define N_      9216
#define M_SORT  16384
#define NT      1024
#define NW      9            // N_/NT ranks owned per thread
#define CLS_THR_  0.6f
#define NMS_THR1_ 0.5f

typedef unsigned long long u64;
typedef unsigned int       u32;

typedef unsigned int tdm_v4u __attribute__((ext_vector_type(4)));
typedef int          tdm_v8i __attribute__((ext_vector_type(8)));
typedef int          tdm_v4i __attribute__((ext_vector_type(4)));

__device__ __forceinline__ void box_from_idx(u32 idx, float& x1, float& y1,
                                             float& x2, float& y2, float& area) {
    const float SCALE = 0.6f;                       // 12.0/20.0 in f32
    float xf = (float)(idx % W_);
    float yf = (float)(idx / W_);
    x1 = rintf((2.0f * xf) / SCALE);                // v_rndne_f32 == round-half-even
    y1 = rintf((2.0f * yf) / SCALE);
    x2 = rintf((2.0f * xf + 12.0f) / SCALE);
    y2 = rintf((2.0f * yf + 12.0f) / SCALE);
    area = (x2 - x1 + 1.0f) * (y2 - y1 + 1.0f);
}

__device__ __forceinline__ float iou_pair(float ax1, float ay1, float ax2, float ay2, float aA,
                                          float bx1, float by1, float bx2, float by2, float bA) {
    float xx1 = fmaxf(ax1, bx1);
    float yy1 = fmaxf(ay1, by1);
    float xx2 = fminf(ax2, bx2);
    float yy2 = fminf(ay2, by2);
    float w = fmaxf(0.0f, xx2 - xx1 + 1.0f);
    float h = fmaxf(0.0f, yy2 - yy1 + 1.0f);
    float inter = w * h;
    return inter / (aA + bA - inter + 1e-10f);
}

extern "C" __global__ __launch_bounds__(NT)
void pnet_detect_kernel(const float* __restrict__ cls,
                        const float* __restrict__ reg,
                        float* __restrict__ out) {
    extern __shared__ unsigned char dynLds[];       // 128 KB dynamic: sort keys
    u64* key = (u64*)dynLds;                        // key[16384]
    __shared__ float         clsLds[N_];            // 36 KB staging (TDM target)
    __shared__ unsigned char sup[N_];               // 9 KB suppression/validity state
    __shared__ u32           rowsLds[32];           // intra-tile IoU rows
    __shared__ u32           sTileKeep;             // kept mask of current tile

    const u32 tid = threadIdx.x;

    // ---------------- Phase 0: TDM DMA cls -> LDS (wave 0 issues) -----------
#if __has_builtin(__builtin_amdgcn_tensor_load_to_lds) && __has_builtin(__builtin_amdgcn_s_wait_tensorcnt)
    if (tid < 32) {
        u32 ldsAddr = (u32)(uintptr_t)&clsLds[0];   // low 32 bits of generic ptr = LDS offset
        unsigned long long ga = (unsigned long long)(uintptr_t)cls;
        // D# group 0: count=1, lds_addr, global_addr[56:0], type=2
        tdm_v4u g0;
        g0.x = 1u;
        g0.y = ldsAddr;
        g0.z = (u32)ga;
        g0.w = ((u32)(ga >> 32) & 0x01FFFFFFu) | 0x80000000u;
        // D# group 1: data_size=4B, tensor 96x96, tile 96x96, stride0=96
        tdm_v8i g1;
        g1.s0 = 0x00020000;        // data_size=2 (4 bytes), no multicast/pad/iterate
        g1.s1 = (W_ << 16);        // tensor_dim0 lo16 in [31:16]
        g1.s2 = (H_ << 16);        // tensor_dim1 lo16 in [31:16]
        g1.s3 = (W_ << 16);        // tile_dim0 in [31:16]
        g1.s4 = H_;                // tile_dim1 in [15:0]
        g1.s5 = W_;                // tensor_dim0_stride lo32
        g1.s6 = (W_ << 16);        // tensor_dim1_stride lo16 in [31:16]
        g1.s7 = 0;
        tdm_v4i gz = {0, 0, 0, 0};
#if __clang_major__ >= 23
        tdm_v8i gz8 = {0, 0, 0, 0, 0, 0, 0, 0};
        __builtin_amdgcn_tensor_load_to_lds(g0, g1, gz, gz, gz8, 0);
#else
        __builtin_amdgcn_tensor_load_to_lds(g0, g1, gz, gz, 0);
#endif
        __builtin_amdgcn_s_wait_tensorcnt(0);
    }
#else
    for (u32 r = tid; r < N_; r += NT) clsLds[r] = cls[r];
#endif
    __syncthreads();

    // ---------------- Phase 1: build descending-sort keys -------------------
    // key = score_bits<<32 | (0xFFFFFFFF - idx): positive f32 bits are order-
    // isomorphic; complemented idx reproduces jnp.argsort's stable tie-break.
    for (u32 r = tid; r < M_SORT; r += NT) {
        u64 k = 0;                                  // padding sorts last
        if (r < N_) {
            u32 bits = __float_as_uint(clsLds[r]);
            k = ((u64)bits << 32) | (u64)(0xFFFFFFFFu - r);
        }
        key[r] = k;
    }
    __syncthreads();

    // ---------------- Phase 2: bitonic descending sort in LDS ---------------
    for (u32 kk = 2; kk <= M_SORT; kk <<= 1) {
        for (u32 j = kk >> 1; j > 0; j >>= 1) {
            for (u32 t = tid; t < M_SORT; t += NT) {
                u32 ixj = t ^ j;
                if (ixj > t) {
                    u64 a = key[t];
                    u64 b = key[ixj];
                    bool desc = ((t & kk) == 0);
                    if ((a < b) == desc) { key[t] = b; key[ixj] = a; }
                }
            }
            __syncthreads();
        }
    }

    // ---------------- Phase 3: per-thread box cache + validity --------------
    float mx1[NW], my1[NW], mx2[NW], my2[NW], mA[NW];
    u32 mySup = 0;                                  // bit s: rank tid+s*NT suppressed/invalid
#pragma unroll
    for (int s = 0; s < NW; ++s) {
        u32 r = tid + (u32)s * NT;
        u64 kr = key[r];
        float sc = __uint_as_float((u32)(kr >> 32));
        u32 idx = 0xFFFFFFFFu - (u32)kr;
        box_from_idx(idx, mx1[s], my1[s], mx2[s], my2[s], mA[s]);
        unsigned char sv = (sc > CLS_THR_) ? 0 : 1; // invalid == pre-suppressed
        if (sv) mySup |= (1u << s);
        sup[r] = sv;
    }
    __syncthreads();

    // ---------------- Phase 4: greedy NMS, 32-rank tiles --------------------
    for (u32 t0 = 0; t0 < N_; t0 += 32) {
        // (a) wave 0 resolves the tile's greedy order internally
        if (tid < 32) {
            u32 l = tid;
            u64 kl = key[t0 + l];
            u32 idl = 0xFFFFFFFFu - (u32)kl;
            float lx1, ly1, lx2, ly2, lA;
            box_from_idx(idl, lx1, ly1, lx2, ly2, lA);
            u32 suppInit = 0, row = 0;
            for (u32 m = 0; m < 32; ++m) {
                u64 km = key[t0 + m];
                u32 idm = 0xFFFFFFFFu - (u32)km;
                float bx1, by1, bx2, by2, bA;
                box_from_idx(idm, bx1, by1, bx2, by2, bA);
                if (sup[t0 + m]) suppInit |= (1u << m);
                float v = iou_pair(lx1, ly1, lx2, ly2, lA, bx1, by1, bx2, by2, bA);
                if (m > l && v > NMS_THR1_) row |= (1u << m);
            }
            rowsLds[l] = row;
            asm volatile("s_wait_dscnt 0" ::: "memory");   // wave-internal LDS RAW
            u32 supp = suppInit, keepm = 0;
            for (u32 i = 0; i < 32; ++i) {
                if (!((supp >> i) & 1u)) { keepm |= (1u << i); supp |= rowsLds[i]; }
            }
            sup[t0 + l] = ((keepm >> l) & 1u) ? 0 : 1;     // finalize tile ranks
            if (l == 0) sTileKeep = keepm;
        }
        __syncthreads();

        // (b) all threads: kept tile members suppress later ranks
        u32 kw = sTileKeep;
        while (kw) {
            u32 i = (u32)__ffs(kw) - 1u;
            kw &= kw - 1u;
            u64 ki = key[t0 + i];
            u32 idi = 0xFFFFFFFFu - (u32)ki;
            float ix1, iy1, ix2, iy2, iA;
            box_from_idx(idi, ix1, iy1, ix2, iy2, iA);
#pragma unroll
            for (int s = 0; s < NW; ++s) {
                u32 j = tid + (u32)s * NT;
                if (j >= t0 + 32 && !((mySup >> s) & 1u)) {
                    float v = iou_pair(ix1, iy1, ix2, iy2, iA,
                                       mx1[s], my1[s], mx2[s], my2[s], mA[s]);
                    if (v > NMS_THR1_) { mySup |= (1u << s); sup[j] = 1; }
                }
            }
        }
        __syncthreads();
    }
    // Reference's second NMS pass (thr=0.7 on keep1) is provably identity:
    // survivors of pass 1 pairwise have IoU <= 0.5 < 0.7. Skipped.

    // ---------------- Phase 5: calibrated output ----------------------------
#pragma unroll
    for (int s = 0; s < NW; ++s) {
        u32 r = tid + (u32)s * NT;
        u64 kr = key[r];
        float sc = __uint_as_float((u32)(kr >> 32));
        u32 idx = 0xFFFFFFFFu - (u32)kr;
        float x1, y1, x2, y2, A;
        box_from_idx(idx, x1, y1, x2, y2, A);
        float keepf = sup[r] ? 0.0f : 1.0f;
        float bw = x2 - x1 + 1.0f;
        float bh = y2 - y1 + 1.0f;
        const float* d = reg + (size_t)idx * 4u;
        float d0 = d[0], d1 = d[1], d2 = d[2], d3 = d[3];
        float* o = out + (size_t)r * 5u;
        o[0] = (x1 + d0 * bw) * keepf;
        o[1] = (y1 + d1 * bh) * keepf;
        o[2] = (x2 + d2 * bw) * keepf;
        o[3] = (y2 + d3 * bh) * keepf;
        o[4] = sc * keepf;
    }
}

extern "C" void kernel_launch(void* const* d_in, const int* in_sizes, int n_in,
                              void* d_out, int out_size, void* d_ws, size_t ws_size,
                              hipStream_t stream) {
    (void)in_sizes; (void)n_in; (void)out_size; (void)d_ws; (void)ws_size;
    const float* cls = (const float*)d_in[0];   // (96,96) f32
    const float* reg = (const float*)d_in[1];   // (96,96,4) f32
    float* out = (float*)d_out;                 // (9216,5) f32
    size_t dynBytes = (size_t)M_SORT * sizeof(unsigned long long);  // 128 KB keys
    hipLaunchKernelGGL(pnet_detect_kernel, dim3(1), dim3(NT), dynBytes, stream,
                       cls, reg, out);
}